// IcoMaxIndexUpSample_8641474199783
// MI455X (gfx1250) — compile-verified
//
#include <hip/hip_runtime.h>
#include <hip/hip_bf16.h>

typedef __attribute__((ext_vector_type(2))) float v2f;
typedef __attribute__((ext_vector_type(4))) float v4f;
typedef __attribute__((ext_vector_type(8))) float v8f;

#define B_SZ   16
#define C_IN   64
#define C_OUT  32
#define V_HI   163842
#define V_LO   40962
#define K_NB   7

#define WAVES_PER_BLOCK 4
#define TILE_N 16
#define N_PER_BLOCK (WAVES_PER_BLOCK * TILE_N)   // 64 vertices per block

// ---------------------------------------------------------------------------
// Zero the (B, C_OUT, V_HI) output (harness poisons it with 0xAA).
// 335.5 MB streamed with non-temporal 128-bit stores (output exceeds the
// 192 MB L2; keep it out of cache so reused index tables stay resident).
// ---------------------------------------------------------------------------
__global__ void zero_out_kernel(float* __restrict__ y, size_t n) {
    size_t n4 = n >> 2;
    v4f* y4 = reinterpret_cast<v4f*>(y);
    size_t stride = (size_t)gridDim.x * blockDim.x;
    const v4f z = {0.f, 0.f, 0.f, 0.f};
    for (size_t i = (size_t)blockIdx.x * blockDim.x + threadIdx.x; i < n4; i += stride) {
        __builtin_nontemporal_store(z, &y4[i]);
    }
    if (blockIdx.x == 0 && threadIdx.x == 0) {
        for (size_t i = n4 << 2; i < n; ++i) y[i] = 0.f;
    }
}

// ---------------------------------------------------------------------------
// Fused GEMM (WMMA f32 16x16x4) + index-gather scatter.
// One wave computes a 32x16 tile of h = W @ x[b] (two 16x16 accumulators,
// shared B fragment), then scatters each element to
//   y[b, o, up_neigh[down[v]*7 + pool[b,o,v]]]
//
// Cache policy: x / pool are single-use streams -> NT loads; y is a
// single-use scatter -> NT stores; W / bias / up_neigh / down are reused
// device-wide -> default RT so they stay in L2.
// ---------------------------------------------------------------------------
__global__ void __launch_bounds__(WAVES_PER_BLOCK * 32)
gemm_scatter_kernel(const float* __restrict__ x,
                    const float* __restrict__ W,
                    const float* __restrict__ bias,
                    const int*   __restrict__ pool,
                    const int*   __restrict__ up_neigh,
                    const int*   __restrict__ down,
                    float*       __restrict__ y) {
    const int lane = threadIdx.x & 31;
    const int wave = threadIdx.x >> 5;

    const int tilesPerBatch = (V_LO + N_PER_BLOCK - 1) / N_PER_BLOCK;  // 641
    const int b    = blockIdx.x / tilesPerBatch;
    const int tile = blockIdx.x % tilesPerBatch;

    const int half = lane >> 4;    // which half-wave (selects K pair / M block)
    const int l16  = lane & 15;    // M (for A) / N (for B, C, D)

    const int vcol  = tile * N_PER_BLOCK + wave * TILE_N + l16;     // this lane's column
    const int vload = vcol < V_LO ? vcol : (V_LO - 1);              // clamped for loads

    const float* xb = x + (size_t)b * C_IN * V_LO;

    v8f acc0 = {};   // output rows o = 0..15
    v8f acc1 = {};   // output rows o = 16..31

    // K loop: 16 steps of K=4  ->  16 x 2 v_wmma_f32_16x16x4_f32 per wave
    #pragma unroll
    for (int k0 = 0; k0 < C_IN; k0 += 4) {
        const int c0 = k0 + 2 * half;   // lane's K pair: c0, c0+1

        // A fragments: A[M=l16][K] = W[o, c]   (row-major W: o*C_IN + c)
        v2f a0, a1, bf;
        const v2f* w0 = reinterpret_cast<const v2f*>(&W[(size_t)l16 * C_IN + c0]);
        const v2f* w1 = reinterpret_cast<const v2f*>(&W[(size_t)(16 + l16) * C_IN + c0]);
        a0 = *w0;
        a1 = *w1;

        // B fragment: B[K][N] = x[b, c, vcol]  (coalesced, single-use -> NT)
        bf[0] = __builtin_nontemporal_load(&xb[(size_t)(c0 + 0) * V_LO + vload]);
        bf[1] = __builtin_nontemporal_load(&xb[(size_t)(c0 + 1) * V_LO + vload]);

        acc0 = __builtin_amdgcn_wmma_f32_16x16x4_f32(false, a0, false, bf,
                                                     (short)0, acc0, false, false);
        acc1 = __builtin_amdgcn_wmma_f32_16x16x4_f32(false, a1, false, bf,
                                                     (short)0, acc1, false, false);
    }

    // Scatter: D element r on this lane is h[b, o = m*16 + r + 8*half, vcol].
    if (vcol < V_LO) {
        const int dv = down[vcol];                       // 0..V_HI-1 (reused table, RT)
        const int*  nb    = up_neigh + (size_t)dv * K_NB;
        float*      yb    = y + (size_t)b * C_OUT * V_HI;
        const int*  poolb = pool + ((size_t)b * C_OUT) * V_LO + vcol;

        #pragma unroll
        for (int r = 0; r < 8; ++r) {
            const int o0 = r + 8 * half;        // accumulator 0 row
            const int o1 = 16 + o0;             // accumulator 1 row

            const int p0 = __builtin_nontemporal_load(&poolb[(size_t)o0 * V_LO]);
            const int p1 = __builtin_nontemporal_load(&poolb[(size_t)o1 * V_LO]);
            const int vert0 = nb[p0];           // 4.6 MB table, reused -> RT
            const int vert1 = nb[p1];

            __builtin_nontemporal_store(acc0[r] + bias[o0], &yb[(size_t)o0 * V_HI + vert0]);
            __builtin_nontemporal_store(acc1[r] + bias[o1], &yb[(size_t)o1 * V_HI + vert1]);
        }
    }
}

// ---------------------------------------------------------------------------
// Launch
// ---------------------------------------------------------------------------
extern "C" void kernel_launch(void* const* d_in, const int* in_sizes, int n_in,
                              void* d_out, int out_size, void* d_ws, size_t ws_size,
                              hipStream_t stream) {
    const float* x    = (const float*)d_in[0];   // (16, 64, 40962)
    const float* W    = (const float*)d_in[1];   // (32, 64)
    const float* bias = (const float*)d_in[2];   // (32,)
    const int*   pool = (const int*)d_in[3];     // (16, 32, 40962)
    const int*   up   = (const int*)d_in[4];     // (163842, 7)
    const int*   down = (const int*)d_in[5];     // (40962,)
    float*       y    = (float*)d_out;           // (16, 32, 163842)

    // 1) zero output (non-temporal stream)
    zero_out_kernel<<<8192, 256, 0, stream>>>(y, (size_t)out_size);

    // 2) fused WMMA GEMM + scatter
    const int tilesPerBatch = (V_LO + N_PER_BLOCK - 1) / N_PER_BLOCK;  // 641
    dim3 grid(B_SZ * tilesPerBatch);                                   // 10256 blocks
    dim3 block(WAVES_PER_BLOCK * 32);                                  // 128 thr = 4 waves
    gemm_scatter_kernel<<<grid, block, 0, stream>>>(x, W, bias, pool, up, down, y);
}